// Quantizer_26268019982518
// MI455X (gfx1250) — compile-verified
//
#include <hip/hip_runtime.h>

#define K_NUM 512
#define SIGMA 1.0f
#define LOG2E 1.44269504088896340736f

typedef float v2f __attribute__((ext_vector_type(2)));
typedef float v8f __attribute__((ext_vector_type(8)));

#if defined(__has_builtin)
#if __has_builtin(__builtin_amdgcn_wmma_f32_16x16x4_f32)
#define HAVE_WMMA_F32X4 1
#endif
#endif

// Wave-tile design (wave32): each wave owns 16 elements (rows M=0..15).
// Loop over 32 chunks of 16 centers. Per chunk one V_WMMA_F32_16X16X4_F32
// computes the 16x16 logit tile
//   L[m][n] = x_m * (2*sigma*log2e*c_n) + 1 * (-sigma*log2e*c_n^2)
// using the K=0 and K=1 rows of the rank-1 product, with C = inline 0
// (no C->D register copies). exp2 -> TRANS pipe; sums/argmax stay in the
// WMMA C/D register layout and are reduced across lanes once at the end.
__global__ __launch_bounds__(256) void quantizer_kernel(
    const float* __restrict__ data,
    const float* __restrict__ centers,
    float* __restrict__ o_zbar,
    float* __restrict__ o_soft,
    float* __restrict__ o_hard,
    int*   __restrict__ o_sym)
{
    __shared__ float s_c[K_NUM];           // c_k
    __shared__ float s_a[K_NUM];           // 2*sigma*log2e*c_k
    __shared__ float s_b[K_NUM];           // -sigma*log2e*c_k^2
    __shared__ float s_stage[8 * 16 * 4];  // per-wave result transpose buffer

    const int tid  = threadIdx.x;
    const int lane = tid & 31;
    const int wave = tid >> 5;
    const int nl   = lane & 15;   // column index within tile
    const int hi   = lane >> 4;   // 0: lanes 0-15 (M=v), 1: lanes 16-31 (M=v+8)

    // Stage centers + derived coefficients (2 per thread).
    for (int i = tid; i < K_NUM; i += 256) {
        float c = centers[i];
        s_c[i] = c;
        s_a[i] = 2.0f * SIGMA * LOG2E * c;
        s_b[i] = -SIGMA * LOG2E * c * c;
    }
    __syncthreads();

    const int eb = blockIdx.x * 128 + wave * 16;  // first element of this wave's tile

    // A matrix (16x4 f32): A[m][0] = x_m, A[m][1] = 1.0, K=2,3 zero.
    // Layout: VGPR0 lanes 0-15 = K=0, VGPR1 lanes 0-15 = K=1; lanes 16-31 = K=2,3.
    float xv = data[eb + nl];
    v2f A;
    A.x = hi ? 0.0f : xv;
    A.y = hi ? 0.0f : 1.0f;

#ifndef HAVE_WMMA_F32X4
    // Fallback needs x_{v+8*hi} broadcast per lane.
    float xm[8];
    #pragma unroll
    for (int v = 0; v < 8; ++v) xm[v] = __shfl(xv, v + 8 * hi, 32);
#endif

    v8f sum_e, sum_ec, best;
    int bidx[8];
    #pragma unroll
    for (int v = 0; v < 8; ++v) {
        sum_e[v] = 0.0f; sum_ec[v] = 0.0f; best[v] = -1e30f; bidx[v] = 0;
    }

    #pragma unroll 4
    for (int t = 0; t < K_NUM / 16; ++t) {
        const int kbase = t * 16;
        float av = s_a[kbase + nl];       // LDS broadcast within half-wave
        float bv = s_b[kbase + nl];
        float cv = s_c[kbase + nl];

        v8f L;
#ifdef HAVE_WMMA_F32X4
        // B matrix (4x16 f32): B[0][n] = a_n, B[1][n] = b_n, K=2,3 zero.
        v2f B;
        B.x = hi ? 0.0f : av;
        B.y = hi ? 0.0f : bv;
        v8f Cz = {};                      // inline SRC2 = 0, no register copies
        L = __builtin_amdgcn_wmma_f32_16x16x4_f32(
                /*neg_a=*/false, A, /*neg_b=*/false, B,
                /*c_mod=*/(short)0, Cz, /*reuse_a=*/false, /*reuse_b=*/false);
#else
        #pragma unroll
        for (int v = 0; v < 8; ++v) L[v] = fmaf(xm[v], av, bv);
#endif

        const int cand = kbase + nl;
        #pragma unroll
        for (int v = 0; v < 8; ++v) {
            float e = __builtin_amdgcn_exp2f(L[v]);   // exp(logit'), base folded in
            sum_e[v]  += e;
            sum_ec[v]  = fmaf(e, cv, sum_ec[v]);
            bool gt = L[v] > best[v];                  // strict > keeps lowest index
            bidx[v] = gt ? cand : bidx[v];
            best[v] = gt ? L[v] : best[v];
        }
    }

    // Cross-lane reduction within each 16-lane column group (xor 1,2,4,8).
    #pragma unroll
    for (int v = 0; v < 8; ++v) {
        float se  = sum_e[v];
        float sec = sum_ec[v];
        float bv  = best[v];
        int   bi  = bidx[v];
        #pragma unroll
        for (int m = 1; m < 16; m <<= 1) {
            se  += __shfl_xor(se,  m, 32);
            sec += __shfl_xor(sec, m, 32);
            float obv = __shfl_xor(bv, m, 32);
            int   obi = __shfl_xor(bi, m, 32);
            bool take = (obv > bv) || ((obv == bv) && (obi < bi));
            bv = take ? obv : bv;
            bi = take ? obi : bi;
        }
        if (nl == 0) {
            int m_row = v + 8 * hi;
            int base  = (wave * 16 + m_row) * 4;
            s_stage[base + 0] = se;
            s_stage[base + 1] = sec;
            s_stage[base + 2] = __int_as_float(bi);
        }
    }
    __syncthreads();

    // Transposed writeout: one element per lane, coalesced 64B stores per wave.
    if (lane < 16) {
        int base  = (wave * 16 + lane) * 4;
        float se  = s_stage[base + 0];
        float sec = s_stage[base + 1];
        int   bi  = __float_as_int(s_stage[base + 2]);
        float soft = sec / se;
        float hard = s_c[bi];
        float zbar = soft + (hard - soft);   // matches reference fp expression
        int g = eb + lane;
        o_zbar[g] = zbar;
        o_soft[g] = soft;
        o_hard[g] = hard;
        o_sym[g]  = bi;
    }
}

extern "C" void kernel_launch(void* const* d_in, const int* in_sizes, int n_in,
                              void* d_out, int out_size, void* d_ws, size_t ws_size,
                              hipStream_t stream) {
    const float* data    = (const float*)d_in[0];
    const float* centers = (const float*)d_in[1];
    const int n = in_sizes[0];            // 524288, divisible by 128

    float* out    = (float*)d_out;        // [zbar | softout | hardout | symbols]
    float* o_zbar = out;
    float* o_soft = out + n;
    float* o_hard = out + 2 * (size_t)n;
    int*   o_sym  = (int*)(out + 3 * (size_t)n);

    const int blocks = n / 128;           // 128 elements per 256-thread block
    quantizer_kernel<<<blocks, 256, 0, stream>>>(data, centers,
                                                 o_zbar, o_soft, o_hard, o_sym);
}